// NollaFraud_26405458936170
// MI455X (gfx1250) — compile-verified
//
#include <hip/hip_runtime.h>
#include <math.h>

typedef __attribute__((ext_vector_type(2))) float v2f;
typedef __attribute__((ext_vector_type(8))) float v8f;

#define N_NODES 50000
#define F_IN    32
#define DEG     32
#define BATCH   2048
#define NREL    3
#define D0      64    // h0 width
#define D1      128   // inter1 width

// ---------------------------------------------------------------------------
// Kernel 1: h0 = relu(feat @ w_mlp + b_mlp) using V_WMMA_F32_16X16X4_F32.
// One wave per 16-row M tile; 4 column tiles of 16 (D0=64); K=32 in 8 steps.
// 50000 rows = 3125 exact tiles -> tile guard is wave-uniform (EXEC all-1s).
//
// f32 WMMA operand layout (per CDNA5 ISA):
//   A 16x4 : lanes 0-15 hold {K=0,K=1} for M=lane; lanes 16-31 hold {K=2,K=3}
//   B 4x16 : VGPR v, lanes 0-15 = B[v][lane]; lanes 16-31 = B[v+2][lane-16]
//   C/D    : VGPR v, lanes 0-15 = C[v][lane];  lanes 16-31 = C[v+8][lane-16]
// ---------------------------------------------------------------------------
__global__ __launch_bounds__(256) void k_mlp_wmma(
    const float* __restrict__ feat,   // (N, 32) row-major
    const float* __restrict__ w,      // (32, 64) row-major
    const float* __restrict__ bias,   // (64,)
    float* __restrict__ h0)           // (N, 64)
{
  const int lane = threadIdx.x & 31;
  const int wave = threadIdx.x >> 5;
  const int tile = blockIdx.x * 8 + wave;         // 16-row tile index
  if (tile >= N_NODES / 16) return;               // uniform per wave
  const int m0   = tile * 16;
  const int half = lane >> 4;                     // 0 or 1
  const int l15  = lane & 15;

  v8f acc0 = {}, acc1 = {}, acc2 = {}, acc3 = {};

  for (int kk = 0; kk < 8; ++kk) {
    const int k = kk * 4;
    // A fragment: row m0+l15, columns k+2*half, k+2*half+1
    v2f a;
    const float* arow = feat + (m0 + l15) * F_IN + k + 2 * half;
    a.x = arow[0];
    a.y = arow[1];
    // B fragments per column tile
    const int kb = k + 2 * half;
    {
      v2f b; b.x = w[(kb + 0) * D0 +  0 + l15]; b.y = w[(kb + 1) * D0 +  0 + l15];
      acc0 = __builtin_amdgcn_wmma_f32_16x16x4_f32(false, a, false, b, (short)0, acc0, false, false);
    }
    {
      v2f b; b.x = w[(kb + 0) * D0 + 16 + l15]; b.y = w[(kb + 1) * D0 + 16 + l15];
      acc1 = __builtin_amdgcn_wmma_f32_16x16x4_f32(false, a, false, b, (short)0, acc1, false, false);
    }
    {
      v2f b; b.x = w[(kb + 0) * D0 + 32 + l15]; b.y = w[(kb + 1) * D0 + 32 + l15];
      acc2 = __builtin_amdgcn_wmma_f32_16x16x4_f32(false, a, false, b, (short)0, acc2, false, false);
    }
    {
      v2f b; b.x = w[(kb + 0) * D0 + 48 + l15]; b.y = w[(kb + 1) * D0 + 48 + l15];
      acc3 = __builtin_amdgcn_wmma_f32_16x16x4_f32(false, a, false, b, (short)0, acc3, false, false);
    }
  }

  // Write back with fused bias + relu. D VGPR v -> row m0+v+8*half, col nt*16+l15.
  #pragma unroll
  for (int v = 0; v < 8; ++v) {
    const int row = m0 + v + 8 * half;
    float* out = h0 + row * D0 + l15;
    out[ 0] = fmaxf(acc0[v] + bias[ 0 + l15], 0.f);
    out[16] = fmaxf(acc1[v] + bias[16 + l15], 0.f);
    out[32] = fmaxf(acc2[v] + bias[32 + l15], 0.f);
    out[48] = fmaxf(acc3[v] + bias[48 + l15], 0.f);
  }
}

// per-row softmax over R=3 logits
__device__ __forceinline__ void softmax3(const float* __restrict__ a, float* w) {
  float a0 = a[0], a1 = a[1], a2 = a[2];
  float m = fmaxf(a0, fmaxf(a1, a2));
  float e0 = __expf(a0 - m), e1 = __expf(a1 - m), e2 = __expf(a2 - m);
  float inv = 1.0f / (e0 + e1 + e2);
  w[0] = e0 * inv; w[1] = e1 * inv; w[2] = e2 * inv;
}

// ---------------------------------------------------------------------------
// Kernel 2: layer-1 InterAgg for ALL nodes.
// Block = 64 threads = one node; thread t owns embedding dim t.
// inter1[n, t]      = sum_r agg_r[t]            * softmax(alpha1[t,:])[r]
// inter1[n, 64+t]   = sum_r (self[t] - agg_r[t]) * softmax(alpha1[64+t,:])[r]
// ---------------------------------------------------------------------------
__global__ __launch_bounds__(64) void k_agg1(
    const float* __restrict__ h0,       // (N, 64)
    const int*   __restrict__ adj,      // (R, N, 32)
    const float* __restrict__ alpha1,   // (128, 3)
    float* __restrict__ inter1)         // (N, 128)
{
  __shared__ int nb[NREL * DEG];        // 96 neighbor ids
  const int n = blockIdx.x;
  const int t = threadIdx.x;

  {
    int i = t;                                   // 0..63
    nb[i] = adj[(i >> 5) * (N_NODES * DEG) + n * DEG + (i & 31)];
    if (t < NREL * DEG - 64) {                   // t < 32 -> 64..95
      int j = t + 64;
      nb[j] = adj[(j >> 5) * (N_NODES * DEG) + n * DEG + (j & 31)];
    }
  }
  __syncthreads();

  const float self = h0[n * D0 + t];
  float wa[3], wb[3];
  softmax3(alpha1 + t * NREL, wa);
  softmax3(alpha1 + (64 + t) * NREL, wb);

  float acc_a = 0.f, acc_b = 0.f;
  #pragma unroll
  for (int r = 0; r < NREL; ++r) {
    float s = 0.f;
    const int* nbr = nb + r * DEG;
    #pragma unroll 8
    for (int k = 0; k < DEG; ++k)
      s += h0[nbr[k] * D0 + t];                  // coalesced across the block
    const float agg = s * (1.0f / DEG);
    acc_a += agg * wa[r];
    acc_b += (self - agg) * wb[r];
  }
  inter1[n * D1 + t]      = acc_a;
  inter1[n * D1 + 64 + t] = acc_b;
}

// ---------------------------------------------------------------------------
// Kernel 3: layer-2 InterAgg (batch nodes only) fused with the 448->2->1 head.
// Block = 128 threads = one batch node; thread e owns embedding dim e.
// h2 layout: [0,64)=h0[n], [64,192)=inter1[n], [192,320)=agg-blend,
//            [320,448)=diff-blend. Head is a length-448 dot with w_d2's 2 cols,
// reduced across the block in LDS.
// ---------------------------------------------------------------------------
__global__ __launch_bounds__(128) void k_agg2_head(
    const int*   __restrict__ nodes,    // (2048,)
    const float* __restrict__ h0,       // (N, 64)
    const float* __restrict__ inter1,   // (N, 128)
    const int*   __restrict__ adj,      // (R, N, 32)
    const float* __restrict__ alpha2,   // (256, 3)
    const float* __restrict__ w_d2,     // (448, 2)
    const float* __restrict__ b_d2,     // (2,)
    const float* __restrict__ w_d1,     // (2, 1)
    const float* __restrict__ b_d1,     // (1,)
    const float* __restrict__ prior,    // (2,)
    float* __restrict__ out)            // (2048,)
{
  __shared__ int   nb[NREL * DEG];
  __shared__ float red0[128];
  __shared__ float red1[128];

  const int b = blockIdx.x;
  const int e = threadIdx.x;
  const int n = nodes[b];

  if (e < NREL * DEG)
    nb[e] = adj[(e >> 5) * (N_NODES * DEG) + n * DEG + (e & 31)];
  __syncthreads();

  const float self2 = inter1[n * D1 + e];
  float wA[3], wB[3];
  softmax3(alpha2 + e * NREL, wA);
  softmax3(alpha2 + (128 + e) * NREL, wB);

  float i2a = 0.f, i2b = 0.f;
  #pragma unroll
  for (int r = 0; r < NREL; ++r) {
    float s = 0.f;
    const int* nbr = nb + r * DEG;
    #pragma unroll 8
    for (int k = 0; k < DEG; ++k)
      s += inter1[nbr[k] * D1 + e];
    const float agg = s * (1.0f / DEG);
    i2a += agg * wA[r];
    i2b += (self2 - agg) * wB[r];
  }

  // partial head dot-products: dims (64+e), (192+e), (320+e), and e (e<64)
  float p0 = self2 * w_d2[(64  + e) * 2 + 0]
           + i2a   * w_d2[(192 + e) * 2 + 0]
           + i2b   * w_d2[(320 + e) * 2 + 0];
  float p1 = self2 * w_d2[(64  + e) * 2 + 1]
           + i2a   * w_d2[(192 + e) * 2 + 1]
           + i2b   * w_d2[(320 + e) * 2 + 1];
  if (e < D0) {
    const float h = h0[n * D0 + e];
    p0 += h * w_d2[e * 2 + 0];
    p1 += h * w_d2[e * 2 + 1];
  }

  red0[e] = p0;
  red1[e] = p1;
  __syncthreads();
  #pragma unroll
  for (int s = 64; s > 0; s >>= 1) {
    if (e < s) { red0[e] += red0[e + s]; red1[e] += red1[e + s]; }
    __syncthreads();
  }

  if (e == 0) {
    float x0 = red0[0] + b_d2[0];
    float x1 = red1[0] + b_d2[1];
    x0 = x0 > 0.f ? x0 : 0.3f * x0;            // leaky_relu(0.3)
    x1 = x1 > 0.f ? x1 : 0.3f * x1;
    x0 += logf(prior[0]);
    x1 += logf(prior[1]);
    const float y = x0 * w_d1[0] + x1 * w_d1[1] + b_d1[0];
    out[b] = 1.0f / (1.0f + expf(-y));
  }
}

extern "C" void kernel_launch(void* const* d_in, const int* in_sizes, int n_in,
                              void* d_out, int out_size, void* d_ws, size_t ws_size,
                              hipStream_t stream) {
  const int*   nodes  = (const int*)  d_in[0];
  const float* feat   = (const float*)d_in[1];
  const int*   adj    = (const int*)  d_in[2];
  const float* prior  = (const float*)d_in[3];
  const float* w_mlp  = (const float*)d_in[4];
  const float* b_mlp  = (const float*)d_in[5];
  const float* alpha1 = (const float*)d_in[6];
  const float* alpha2 = (const float*)d_in[7];
  const float* w_d2   = (const float*)d_in[8];
  const float* b_d2   = (const float*)d_in[9];
  const float* w_d1   = (const float*)d_in[10];
  const float* b_d1   = (const float*)d_in[11];
  float* out = (float*)d_out;

  // workspace: h0 (N x 64) then inter1 (N x 128), 38.4 MB total
  float* h0     = (float*)d_ws;
  float* inter1 = h0 + (size_t)N_NODES * D0;

  // 3125 M-tiles, 8 waves (256 threads) per block
  const int k1_blocks = (N_NODES / 16 + 7) / 8;   // 391
  k_mlp_wmma<<<k1_blocks, 256, 0, stream>>>(feat, w_mlp, b_mlp, h0);

  k_agg1<<<N_NODES, 64, 0, stream>>>(h0, adj, alpha1, inter1);

  k_agg2_head<<<BATCH, 128, 0, stream>>>(nodes, h0, inter1, adj, alpha2,
                                         w_d2, b_d2, w_d1, b_d1, prior, out);
}